// Krylov_61435212202077
// MI455X (gfx1250) — compile-verified
//
#include <hip/hip_runtime.h>

// Krylov basis on MI455X (gfx1250): out[n, k, f] = (A^k b)[n, f]
// 31 sequential fp32 WMMA GEMMs (v_wmma_f32_16x16x4_f32) + one copy kernel.
//
// GEMM: 64x64 block tile, 128 threads = 4 wave32 in a 2x2 grid; each wave
// computes a 32x32 C-tile as 2x2 WMMA subtiles (4x v8f accumulators).
// A tile row-major in LDS (stride 36); B tile TRANSPOSED in LDS (stride 34)
// so both WMMA operands are contiguous 8-byte ds loads (no repack movs).

typedef __attribute__((ext_vector_type(2))) float v2f;
typedef __attribute__((ext_vector_type(4))) float v4f;
typedef __attribute__((ext_vector_type(8))) float v8f;

#define N_DIM 4096
#define F_DIM 256
#define KDIMS 32

#define BM 64
#define BN 64
#define BK 32
#define LDA 36   // A tile LDS stride (floats): mult of 4 (b128 stores), even (b64 reads), conflict-free
#define LDBT 34  // transposed-B tile LDS stride (floats): even (b64 reads), conflict-free reads

// ---------------------------------------------------------------------------
// Step 0: out[:, 0, :] = b
// ---------------------------------------------------------------------------
__global__ __launch_bounds__(256) void krylov_copy_b(const float* __restrict__ b,
                                                     float* __restrict__ out) {
    int idx = blockIdx.x * 256 + threadIdx.x;   // float4 index, 4096*64 total
    int n  = idx >> 6;                          // 64 float4 per row of F=256
    int f4 = idx & 63;
    const v4f* src = (const v4f*)(b + (size_t)n * F_DIM) + f4;
    v4f*       dst = (v4f*)(out + (size_t)n * (KDIMS * F_DIM)) + f4;
    *dst = *src;
}

// ---------------------------------------------------------------------------
__global__ __launch_bounds__(128) void krylov_gemm(
    const float* __restrict__ A,
    const float* __restrict__ Vin, int vin_stride,
    float* __restrict__ Vout, int vout_stride)
{
    __shared__ __align__(16) float As[BM * LDA];
    __shared__ __align__(16) float Bst[BN * LDBT];   // [n][k] transposed

    const int tid  = threadIdx.x;
    const int lane = tid & 31;
    const int wave = tid >> 5;                 // 0..3

    const int block_m = blockIdx.x * BM;
    const int block_n = blockIdx.y * BN;

    // wave -> 32x32 C-subtile
    const int wm = (wave >> 1) * 32;
    const int wn = (wave & 1) * 32;

    // WMMA f32 16x16x4 lane mapping (ISA 7.12.2):
    //  A: lanes 0-15 -> M=lane, VGPRs=K{0,1}; lanes 16-31 -> M=lane-16, K{2,3}
    //  B: mirrored with lane index = N
    //  D: VGPR r -> M = r + 8*(lane>>4), N = lane&15
    const int half = lane >> 4;                // 0 or 1
    const int mn   = lane & 15;                // M (A) / N (B,D) within 16
    const int kb   = half * 2;                 // K-pair base

    // global->LDS loader mapping (128 threads, 4 rounds each matrix)
    const int row_a = tid >> 3;                // 0..15 (+16 per round)
    const int col_a = (tid & 7) * 4;           // 0..28
    const int row_b = tid >> 4;                // 0..7  (+8 per round)
    const int col_b = (tid & 15) * 4;          // 0..60

    const float* ag = A   + (size_t)(block_m + row_a) * N_DIM + col_a;
    const float* bg = Vin + (size_t)row_b * vin_stride + block_n + col_b;

    v8f acc00 = {}, acc01 = {}, acc10 = {}, acc11 = {};

    // ---- preload tile k0 = 0 ----
    v4f ra[4], rb[4];
    #pragma unroll
    for (int r = 0; r < 4; ++r) {
        ra[r] = *(const v4f*)(ag + (size_t)(16 * r) * N_DIM);
        rb[r] = *(const v4f*)(bg + (size_t)(8 * r) * vin_stride);
    }

    for (int k0 = 0; k0 < N_DIM; k0 += BK) {
        __syncthreads();   // previous iteration's LDS reads complete
        #pragma unroll
        for (int r = 0; r < 4; ++r) {
            *(v4f*)&As[(row_a + 16 * r) * LDA + col_a] = ra[r];
            // transposed scatter of B: Bst[n][k]
            #pragma unroll
            for (int j = 0; j < 4; ++j)
                Bst[(col_b + j) * LDBT + row_b + 8 * r] = rb[r][j];
        }
        __syncthreads();

        // ---- issue next tile's global loads; they retire under the WMMAs ----
        if (k0 + BK < N_DIM) {
            const float* agn = ag + (k0 + BK);
            const float* bgn = bg + (size_t)(k0 + BK) * vin_stride;
            #pragma unroll
            for (int r = 0; r < 4; ++r) {
                ra[r] = *(const v4f*)(agn + (size_t)(16 * r) * N_DIM);
                rb[r] = *(const v4f*)(bgn + (size_t)(8 * r) * vin_stride);
            }
            __builtin_prefetch(agn + BK, 0, 1);   // global_prefetch_b8 for tile k0+64
        }

        // ---- 8 K-steps of 4; 2x2 subtiles -> 32 WMMAs per tile ----
        const float* pa0 = &As[(wm + mn) * LDA + kb];
        const float* pa1 = &As[(wm + 16 + mn) * LDA + kb];
        const float* pb0 = &Bst[(wn + mn) * LDBT + kb];
        const float* pb1 = &Bst[(wn + 16 + mn) * LDBT + kb];
        #pragma unroll
        for (int kk = 0; kk < BK; kk += 4) {
            v2f a0 = *(const v2f*)(pa0 + kk);
            v2f a1 = *(const v2f*)(pa1 + kk);
            v2f b0 = *(const v2f*)(pb0 + kk);
            v2f b1 = *(const v2f*)(pb1 + kk);
            acc00 = __builtin_amdgcn_wmma_f32_16x16x4_f32(
                false, a0, false, b0, (short)0, acc00, false, false);
            acc01 = __builtin_amdgcn_wmma_f32_16x16x4_f32(
                false, a0, false, b1, (short)0, acc01, false, false);
            acc10 = __builtin_amdgcn_wmma_f32_16x16x4_f32(
                false, a1, false, b0, (short)0, acc10, false, false);
            acc11 = __builtin_amdgcn_wmma_f32_16x16x4_f32(
                false, a1, false, b1, (short)0, acc11, false, false);
        }
    }

    // ---- store D tiles (strided output layout) ----
    float* o = Vout + (size_t)(block_m + wm + half * 8) * vout_stride
                    + block_n + wn + mn;
    #pragma unroll
    for (int r = 0; r < 8; ++r) {
        o[(size_t)r * vout_stride]             = acc00[r];
        o[(size_t)r * vout_stride + 16]        = acc01[r];
        o[(size_t)(r + 16) * vout_stride]      = acc10[r];
        o[(size_t)(r + 16) * vout_stride + 16] = acc11[r];
    }
}

// ---------------------------------------------------------------------------
extern "C" void kernel_launch(void* const* d_in, const int* in_sizes, int n_in,
                              void* d_out, int out_size, void* d_ws, size_t ws_size,
                              hipStream_t stream) {
    const float* A = (const float*)d_in[0];   // [4096, 4096]
    const float* b = (const float*)d_in[1];   // [4096, 256]
    float* out = (float*)d_out;               // [4096, 32, 256]

    (void)in_sizes; (void)n_in; (void)out_size; (void)d_ws; (void)ws_size;

    const int row_stride = KDIMS * F_DIM;     // 8192 floats between rows of a slot

    // slot 0 = b
    krylov_copy_b<<<(N_DIM * (F_DIM / 4)) / 256, 256, 0, stream>>>(b, out);

    // slots 1..31: v_k = A @ v_{k-1}, read/write directly in strided layout
    dim3 grid(N_DIM / BM, F_DIM / BN);        // 64 x 4 = 256 blocks
    for (int k = 1; k < KDIMS; ++k) {
        const float* vin = out + (size_t)(k - 1) * F_DIM;
        float*      vout = out + (size_t)k * F_DIM;
        krylov_gemm<<<grid, 128, 0, stream>>>(A, vin, row_stride, vout, row_stride);
    }
}